// Net_89867895701532
// MI455X (gfx1250) — compile-verified
//
#include <hip/hip_runtime.h>

// ---------------------------------------------------------------------------
// Spatial-transformer pipeline for gfx1250 (MI455X, wave32).
//  - conv1 5x5 (4->7) + relu + maxpool2   (direct fp32, float4 channel loads)
//  - conv2 3x3 (7->9) + relu + maxpool2   (direct fp32)
//  - fc1 [4x142884]x[142884x32]           TDM (tensor_load_to_lds) -> LDS ->
//                                         V_WMMA_F32_16X16X4_F32, split-K
//  - fc2/fc3 + R*R^T + translations       (tiny, one block)
//  - fused bilinear warp + alpha composite over 11 layers
// ---------------------------------------------------------------------------

typedef float v2f __attribute__((ext_vector_type(2)));
typedef float v8f __attribute__((ext_vector_type(8)));
typedef unsigned int u32x4 __attribute__((ext_vector_type(4)));
typedef int i32x4 __attribute__((ext_vector_type(4)));
typedef int i32x8 __attribute__((ext_vector_type(8)));

#define SZ     512
#define NP     11
#define BATCH  4
#define ROWS   ((NP + 1) * SZ)      // 6144 rows in the NHWC input
#define P1     254                  // conv1 pooled size
#define P2     126                  // conv2 pooled size
#define FEAT   (9 * 126 * 126)      // 142884

#define OFF_EYE (BATCH * 4 * SZ * SZ)      // 4194304
#define OFF_TR  (OFF_EYE + BATCH * NP * 4) // 4194480

// workspace layout (float offsets)
#define WS_C1    0
#define WS_C2    (WS_C1 + BATCH * 7 * P1 * P1)   // 1,806,448
#define WS_ACC   (WS_C2 + BATCH * FEAT)          // 2,377,984
#define WS_THETA (WS_ACC + BATCH * 32)           // 2,378,112  (+264)

#define KCHUNK   256
#define LDSPITCH 258   // 256 cols + 2 pad DWORDs inserted by TDM pad feature

// ------------------------------- conv1 -------------------------------------
__global__ void conv1_pool(const float4* __restrict__ x4,
                           const float*  __restrict__ w1,
                           const float*  __restrict__ b1,
                           float*        __restrict__ c1) {
    __shared__ float wl[700];
    __shared__ float bl[7];
    for (int i = threadIdx.x; i < 700; i += blockDim.x) wl[i] = w1[i];
    if (threadIdx.x < 7) bl[threadIdx.x] = b1[threadIdx.x];
    __syncthreads();

    int t = blockIdx.x * blockDim.x + threadIdx.x;
    if (t >= BATCH * P1 * P1) return;
    int pw = t % P1, ph = (t / P1) % P1, b = t / (P1 * P1);

    float best[7];
    #pragma unroll
    for (int oc = 0; oc < 7; ++oc) best[oc] = 0.f;  // relu outputs are >= 0

    for (int dy = 0; dy < 2; ++dy)
    for (int dx = 0; dx < 2; ++dx) {
        int oy = 2 * ph + dy, ox = 2 * pw + dx;
        float a[7];
        #pragma unroll
        for (int oc = 0; oc < 7; ++oc) a[oc] = bl[oc];
        for (int ky = 0; ky < 5; ++ky)
        for (int kx = 0; kx < 5; ++kx) {
            float4 px = x4[((long)b * ROWS + (oy + ky)) * SZ + (ox + kx)];
            int wi = ky * 5 + kx;
            #pragma unroll
            for (int oc = 0; oc < 7; ++oc) {
                const float* w = &wl[oc * 100 + wi];
                a[oc] += px.x * w[0] + px.y * w[25] + px.z * w[50] + px.w * w[75];
            }
        }
        #pragma unroll
        for (int oc = 0; oc < 7; ++oc) {
            float r = a[oc] > 0.f ? a[oc] : 0.f;
            if (r > best[oc]) best[oc] = r;
        }
    }
    #pragma unroll
    for (int oc = 0; oc < 7; ++oc)
        c1[((long)(b * 7 + oc) * P1 + ph) * P1 + pw] = best[oc];
}

// ------------------------------- conv2 -------------------------------------
__global__ void conv2_pool(const float* __restrict__ c1,
                           const float* __restrict__ w2,
                           const float* __restrict__ b2,
                           float*       __restrict__ h) {
    __shared__ float wl[567];
    __shared__ float bl[9];
    for (int i = threadIdx.x; i < 567; i += blockDim.x) wl[i] = w2[i];
    if (threadIdx.x < 9) bl[threadIdx.x] = b2[threadIdx.x];
    __syncthreads();

    int t = blockIdx.x * blockDim.x + threadIdx.x;
    if (t >= BATCH * P2 * P2) return;
    int pw = t % P2, ph = (t / P2) % P2, b = t / (P2 * P2);

    float best[9];
    #pragma unroll
    for (int oc = 0; oc < 9; ++oc) best[oc] = 0.f;

    for (int dy = 0; dy < 2; ++dy)
    for (int dx = 0; dx < 2; ++dx) {
        int oy = 2 * ph + dy, ox = 2 * pw + dx;
        float a[9];
        #pragma unroll
        for (int oc = 0; oc < 9; ++oc) a[oc] = bl[oc];
        for (int ic = 0; ic < 7; ++ic)
        for (int ky = 0; ky < 3; ++ky)
        for (int kx = 0; kx < 3; ++kx) {
            float v = c1[((long)(b * 7 + ic) * P1 + oy + ky) * P1 + ox + kx];
            int wi = ic * 9 + ky * 3 + kx;
            #pragma unroll
            for (int oc = 0; oc < 9; ++oc) a[oc] += v * wl[oc * 63 + wi];
        }
        #pragma unroll
        for (int oc = 0; oc < 9; ++oc) {
            float r = a[oc] > 0.f ? a[oc] : 0.f;
            if (r > best[oc]) best[oc] = r;
        }
    }
    // write in flattened [B, 9*126*126] order (NCHW flatten)
    #pragma unroll
    for (int oc = 0; oc < 9; ++oc)
        h[(long)b * FEAT + oc * (P2 * P2) + ph * P2 + pw] = best[oc];
}

// ------------------------------ fc1 (TDM + WMMA) ---------------------------
__global__ void zero_acc(float* acc) {
    if (threadIdx.x < BATCH * 32) acc[threadIdx.x] = 0.f;
}

// Issue a TDM 2D tile load: rows x KCHUNK fp32 elements, row stride
// stride_elems, from gptr into LDS at lds_byte_addr.  LDS rows are padded by
// the TDM pad feature (+2 DWORDs per 256 DWORDs) -> pitch LDSPITCH floats,
// which de-conflicts the 64-bank LDS for the per-lane row reads below.
__device__ __forceinline__ void tdm_load_2d(const float* gptr,
                                            unsigned lds_byte_addr,
                                            int rows, long stride_elems) {
    unsigned long long ga = (unsigned long long)(size_t)gptr;
    u32x4 g0;
    g0[0] = 1u;                                            // count=1, user D#
    g0[1] = lds_byte_addr;                                 // lds_addr [63:32]
    g0[2] = (unsigned)(ga & 0xFFFFFFFFu);                  // global_addr lo
    g0[3] = (unsigned)((ga >> 32) & 0x01FFFFFFu) | (2u << 30); // addr hi | type=2
    i32x8 g1;
    // data_size=2 (4B) | pad_enable | pad_interval=7 (256 dw) | pad_amount=1 (2 dw)
    g1[0] = (2 << 16) | (1 << 20) | (7 << 22) | (1 << 25);
    g1[1] = (KCHUNK & 0xFFFF) << 16;                       // tensor_dim0[15:0] @bit48
    g1[2] = (KCHUNK >> 16) | ((rows & 0xFFFF) << 16);      // dim0 hi | tensor_dim1 lo
    g1[3] = (KCHUNK << 16);                                // tensor_dim1 hi(0) | tile_dim0
    g1[4] = rows;                                          // tile_dim1 | tile_dim2=0
    g1[5] = (int)(stride_elems & 0xFFFFFFFF);              // tensor_dim0_stride lo32
    g1[6] = (int)((stride_elems >> 32) & 0xFFFF);          // stride hi16 | dim1_stride=0
    g1[7] = 0;
    i32x4 z = {0, 0, 0, 0};
#if defined(__clang_major__) && (__clang_major__ >= 23)
    i32x8 z8 = {0, 0, 0, 0, 0, 0, 0, 0};
    __builtin_amdgcn_tensor_load_to_lds(g0, g1, z, z, z8, 0);
#else
    __builtin_amdgcn_tensor_load_to_lds(g0, g1, z, z, 0);
#endif
}

// out[m][n] = sum_k h[m][k] * W[n][k];  M=4 (padded to 16), N=32, K=142884.
// A 16x4 f32 layout: lanes 0-15 -> M=lane, vgpr{0,1}=K{0,1}; lanes 16-31 -> K{2,3}.
// B 4x16 f32 layout (mirror): lanes 0-15 -> N=lane, vgpr{0,1}=K{0,1}; hi lanes K{2,3}.
__global__ void __launch_bounds__(32) fc1_wmma(const float* __restrict__ h,
                                               const float* __restrict__ W,
                                               float*       __restrict__ acc) {
    __shared__ float Wt[32 * LDSPITCH];     // 33,024 B
    __shared__ float Ht[BATCH * LDSPITCH];  //  4,128 B
    int lane  = threadIdx.x;
    int m     = lane & 15;
    int khalf = (lane >> 4) * 2;            // 0 or 2
    long k0   = (long)blockIdx.x * KCHUNK;

    v8f c0 = {};                            // N tile 0..15
    v8f c1 = {};                            // N tile 16..31

    if (k0 + KCHUNK <= FEAT) {
        // -------- main path: TDM-staged, fully coalesced weight stream -----
        tdm_load_2d(W + k0, (unsigned)(size_t)(void*)Wt, 32, (long)FEAT);
        tdm_load_2d(h + k0, (unsigned)(size_t)(void*)Ht, BATCH, (long)FEAT);
        __builtin_amdgcn_s_wait_tensorcnt(0);   // single-wave block: no barrier

        #pragma unroll 4
        for (int ks = 0; ks < KCHUNK / 4; ++ks) {
            int kl = ks * 4 + khalf;
            v2f a = {0.f, 0.f};
            if (m < BATCH) a = *(const v2f*)&Ht[m * LDSPITCH + kl];
            v2f b0 = *(const v2f*)&Wt[m * LDSPITCH + kl];
            v2f b1 = *(const v2f*)&Wt[(16 + m) * LDSPITCH + kl];
            c0 = __builtin_amdgcn_wmma_f32_16x16x4_f32(false, a, false, b0,
                                                       (short)0, c0, false, false);
            c1 = __builtin_amdgcn_wmma_f32_16x16x4_f32(false, a, false, b1,
                                                       (short)0, c1, false, false);
        }
    } else {
        // -------- tail path: last partial chunk (K remainder = 36) ---------
        for (int ks = 0; ks < KCHUNK / 4; ++ks) {
            long k = k0 + (long)ks * 4;
            if (k >= FEAT) break;           // FEAT % 4 == 0, uniform exit
            v2f a = {0.f, 0.f};
            if (m < BATCH)
                a = *(const v2f*)(h + (long)m * FEAT + k + khalf);
            v2f b0 = *(const v2f*)(W + (long)m        * FEAT + k + khalf);
            v2f b1 = *(const v2f*)(W + (long)(16 + m) * FEAT + k + khalf);
            __builtin_prefetch(W + (long)m * FEAT + k + khalf + 64, 0, 1);
            c0 = __builtin_amdgcn_wmma_f32_16x16x4_f32(false, a, false, b0,
                                                       (short)0, c0, false, false);
            c1 = __builtin_amdgcn_wmma_f32_16x16x4_f32(false, a, false, b1,
                                                       (short)0, c1, false, false);
        }
    }

    // C layout: lanes 0-15, vgpr r -> (M=r, N=lane). Only rows 0..3 are real.
    if (lane < 16) {
        #pragma unroll
        for (int r = 0; r < BATCH; ++r) {
            atomicAdd(&acc[r * 32 + lane],      c0[r]);
            atomicAdd(&acc[r * 32 + 16 + lane], c1[r]);
        }
    }
}

// --------------------------- fc2 / fc3 / side outputs ----------------------
__global__ void fc23_theta(const float* __restrict__ acc,
                           const float* __restrict__ fc1_b,
                           const float* __restrict__ fc2_w,
                           const float* __restrict__ fc2_b,
                           const float* __restrict__ fc3_w,
                           const float* __restrict__ fc3_b,
                           float* __restrict__ theta_ws,
                           float* __restrict__ out) {
    __shared__ float h1[BATCH * 32];
    __shared__ float h2[BATCH * 33];
    __shared__ float th[BATCH * 66];
    int t = threadIdx.x;

    if (t < BATCH * 32) h1[t] = fmaxf(acc[t] + fc1_b[t & 31], 0.f);
    __syncthreads();

    for (int idx = t; idx < BATCH * 33; idx += blockDim.x) {
        int b = idx / 33, j = idx % 33;
        float s = fc2_b[j];
        for (int n = 0; n < 32; ++n) s += h1[b * 32 + n] * fc2_w[j * 32 + n];
        h2[idx] = fmaxf(s, 0.f);
    }
    __syncthreads();

    for (int idx = t; idx < BATCH * 66; idx += blockDim.x) {
        int b = idx / 66, o = idx % 66;
        float s = fc3_b[o];
        for (int j = 0; j < 33; ++j) s += h2[b * 33 + j] * fc3_w[o * 33 + j];
        th[idx] = s;
        theta_ws[idx] = s;
    }
    __syncthreads();

    // close_to_eye[b,p,i*2+k] = sum_j R[b,p,i,j] * R[b,p,k,j]
    for (int idx = t; idx < BATCH * NP * 4; idx += blockDim.x) {
        int b = idx / (NP * 4), rem = idx % (NP * 4);
        int p = rem / 4, e = rem % 4, i = e >> 1, kk = e & 1;
        const float* T = &th[b * 66 + p * 6];
        out[OFF_EYE + idx] = T[i * 3 + 0] * T[kk * 3 + 0] + T[i * 3 + 1] * T[kk * 3 + 1];
    }
    // translations[b,p,i] = theta[b,p,i,2]
    for (int idx = t; idx < BATCH * NP * 2; idx += blockDim.x) {
        int b = idx / (NP * 2), rem = idx % (NP * 2);
        int p = rem / 2, i = rem % 2;
        out[OFF_TR + idx] = th[b * 66 + p * 6 + i * 3 + 2];
    }
}

// -------------------- fused bilinear warp + alpha composite ----------------
__global__ void warp_composite(const float4* __restrict__ x4,
                               const float*  __restrict__ theta_ws,
                               float*        __restrict__ out) {
    __shared__ float th[66];
    int t = blockIdx.x * blockDim.x + threadIdx.x;
    int b = t >> 18;                 // 512*512 = 2^18 pixels per batch
    int hh = (t >> 9) & (SZ - 1);
    int ww = t & (SZ - 1);
    if (threadIdx.x < 66) th[threadIdx.x] = theta_ws[b * 66 + threadIdx.x];
    __syncthreads();

    float xs = ((2.f * ww + 1.f) / (float)SZ) - 1.f;
    float ys = ((2.f * hh + 1.f) / (float)SZ) - 1.f;

    float st0 = 0.f, st1 = 0.f, st2 = 0.f, st3 = 0.f;
    for (int p = 0; p < NP; ++p) {
        const float* T = &th[p * 6];
        float gx = T[0] * xs + T[1] * ys + T[2];
        float gy = T[3] * xs + T[4] * ys + T[5];
        float ix = ((gx + 1.f) * SZ - 1.f) * 0.5f;
        float iy = ((gy + 1.f) * SZ - 1.f) * 0.5f;
        float fx = floorf(ix), fy = floorf(iy);
        float wx = ix - fx, wy = iy - fy;
        int x0 = (int)fx, y0 = (int)fy;
        int x0i = min(max(x0, 0), SZ - 1), x1i = min(max(x0 + 1, 0), SZ - 1);
        int y0i = min(max(y0, 0), SZ - 1), y1i = min(max(y0 + 1, 0), SZ - 1);
        long base = (long)b * ROWS + (long)(p + 1) * SZ;
        float4 v00 = x4[(base + y0i) * SZ + x0i];
        float4 v01 = x4[(base + y0i) * SZ + x1i];
        float4 v10 = x4[(base + y1i) * SZ + x0i];
        float4 v11 = x4[(base + y1i) * SZ + x1i];
        float w00 = (1.f - wx) * (1.f - wy), w01 = wx * (1.f - wy);
        float w10 = (1.f - wx) * wy,         w11 = wx * wy;
        float l0 = v00.x * w00 + v01.x * w01 + v10.x * w10 + v11.x * w11;
        float l1 = v00.y * w00 + v01.y * w01 + v10.y * w10 + v11.y * w11;
        float l2 = v00.z * w00 + v01.z * w01 + v10.z * w10 + v11.z * w11;
        float l3 = v00.w * w00 + v01.w * w01 + v10.w * w10 + v11.w * w11;
        if (p == 0) { st0 = l0; st1 = l1; st2 = l2; st3 = l3; }
        else {
            float a = fminf(fmaxf(l3, 0.f), 1.f);
            st0 = fminf(fmaxf(st0 - 2.f * a, -1.f), 1.f);
            st1 = fminf(fmaxf(st1 - 2.f * a, -1.f), 1.f);
            st2 = fminf(fmaxf(st2 - 2.f * a, -1.f), 1.f);
            st3 = fminf(fmaxf(st3 - 2.f * a, -1.f), 1.f);
            st0 = fminf(fmaxf(st0 + l0 + 1.f, -1.f), 1.f);
            st1 = fminf(fmaxf(st1 + l1 + 1.f, -1.f), 1.f);
            st2 = fminf(fmaxf(st2 + l2 + 1.f, -1.f), 1.f);
            st3 = fminf(fmaxf(st3 + l3 + 1.f, -1.f), 1.f);
        }
    }
    long ob = (long)b * 4 * SZ * SZ + (long)hh * SZ + ww;
    out[ob]                      = st0;
    out[ob + (long)SZ * SZ]      = st1;
    out[ob + 2L * SZ * SZ]       = st2;
    out[ob + 3L * SZ * SZ]       = st3;
}

// ------------------------------- launcher ----------------------------------
extern "C" void kernel_launch(void* const* d_in, const int* in_sizes, int n_in,
                              void* d_out, int out_size, void* d_ws, size_t ws_size,
                              hipStream_t stream) {
    (void)in_sizes; (void)n_in; (void)out_size; (void)ws_size;

    const float* x     = (const float*)d_in[0];
    const float* w1    = (const float*)d_in[1];
    const float* b1    = (const float*)d_in[2];
    const float* w2    = (const float*)d_in[3];
    const float* b2    = (const float*)d_in[4];
    const float* fc1_w = (const float*)d_in[5];
    const float* fc1_b = (const float*)d_in[6];
    const float* fc2_w = (const float*)d_in[7];
    const float* fc2_b = (const float*)d_in[8];
    const float* fc3_w = (const float*)d_in[9];
    const float* fc3_b = (const float*)d_in[10];

    float* ws    = (float*)d_ws;
    float* c1    = ws + WS_C1;
    float* c2    = ws + WS_C2;
    float* acc   = ws + WS_ACC;
    float* theta = ws + WS_THETA;
    float* out   = (float*)d_out;

    conv1_pool<<<(BATCH * P1 * P1 + 255) / 256, 256, 0, stream>>>(
        (const float4*)x, w1, b1, c1);
    conv2_pool<<<(BATCH * P2 * P2 + 255) / 256, 256, 0, stream>>>(
        c1, w2, b2, c2);
    zero_acc<<<1, 128, 0, stream>>>(acc);
    fc1_wmma<<<(FEAT + KCHUNK - 1) / KCHUNK, 32, 0, stream>>>(c2, fc1_w, acc);
    fc23_theta<<<1, 128, 0, stream>>>(acc, fc1_b, fc2_w, fc2_b, fc3_w, fc3_b,
                                      theta, out);
    warp_composite<<<(BATCH * SZ * SZ) / 256, 256, 0, stream>>>(
        (const float4*)x, theta, out);
}